// PredictNet_343597384499
// MI455X (gfx1250) — compile-verified
//
#include <hip/hip_runtime.h>

typedef __attribute__((ext_vector_type(2))) float v2f;
typedef __attribute__((ext_vector_type(8))) float v8f;

namespace {
constexpr int kT  = 1024;
constexpr int kB  = 8192;
constexpr int kIn = 3;
constexpr int kH  = 3;
constexpr int kL  = 4;
constexpr int kG  = 4 * kH;   // 12 gate rows
}

// Native gfx1250 V_TANH_F32 if the toolchain exposes it; exp/rcp fallback.
__device__ __forceinline__ float tanh_fast(float x) {
#if __has_builtin(__builtin_amdgcn_tanhf)
    return __builtin_amdgcn_tanhf(x);
#elif __has_builtin(__builtin_amdgcn_tanh_f32)
    return __builtin_amdgcn_tanh_f32(x);
#else
    return 2.0f * __builtin_amdgcn_rcpf(1.0f + __expf(-2.0f * x)) - 1.0f;
#endif
}
__device__ __forceinline__ float sigm(float x) {
#if __has_builtin(__builtin_amdgcn_tanhf) || __has_builtin(__builtin_amdgcn_tanh_f32)
    // sigmoid(x) = 0.5 * tanh(x/2) + 0.5  -> 1 trans + 1 fma
    return __builtin_fmaf(tanh_fast(0.5f * x), 0.5f, 0.5f);
#else
    return __builtin_amdgcn_rcpf(1.0f + __expf(-x));
#endif
}

// One wave32 owns a 16-row batch tile. Weights live as WMMA A-operands,
// bias lives in the WMMA accumulator, h/c state lives in low-lane registers.
__global__ __launch_bounds__(256, 1) void lstm4_wmma(
    const float* __restrict__ xin,  // [T,B,IN]
    const float* __restrict__ w_ih, // [L,4H,IN]
    const float* __restrict__ w_hh, // [L,4H,H]
    const float* __restrict__ b_ih, // [L,4H]
    const float* __restrict__ b_hh, // [L,4H]
    const float* __restrict__ w_fc, // [H,H]
    const float* __restrict__ b_fc, // [H]
    float* __restrict__ out)        // [B,H]
{
    const int lane = threadIdx.x & 31;
    const int wave = threadIdx.x >> 5;
    const int tile = blockIdx.x * (blockDim.x >> 5) + wave;  // 0..511
    const int base = tile * 16;
    const bool lo  = lane < 16;
    const int col  = lane & 15;        // A: gate row M ; B/D: batch column N
    const int kb   = lo ? 0 : 2;       // K pair held by this half-wave

    // ---- per-layer constants: A operands (weights) + bias accumulator ----
    v2f Aih[kL], Ahh[kL];
    v8f Cb[kL];
    v2f Bh[kL];                        // recurrent h in B-operand layout
#pragma unroll
    for (int l = 0; l < kL; ++l) {
        const float* wi = w_ih + l * kG * kIn;
        const float* wh = w_hh + l * kG * kH;
        const bool mv = col < kG;      // gate rows 12..15 are zero padding
        v2f a;                          // A 16x4: lanes 0-15 K=0,1 ; 16-31 K=2,3
        a.x = mv ? wi[col * kIn + kb] : 0.0f;
        a.y = (mv && lo) ? wi[col * kIn + 1] : 0.0f;   // K=3 is zero pad
        Aih[l] = a;
        v2f b;
        b.x = mv ? wh[col * kH + kb] : 0.0f;
        b.y = (mv && lo) ? wh[col * kH + 1] : 0.0f;
        Ahh[l] = b;
        v8f cb;                         // D/C layout: vgpr r = gate r (+8 hi half)
#pragma unroll
        for (int r = 0; r < 8; ++r) {
            int gi = r + (lo ? 0 : 8);
            cb[r] = (gi < kG) ? (b_ih[l * kG + gi] + b_hh[l * kG + gi]) : 0.0f;
        }
        Cb[l] = cb;
        v2f z; z.x = 0.0f; z.y = 0.0f;
        Bh[l] = z;                      // h0 = 0
    }

    float hs[kL][kH], cs[kL][kH];
#pragma unroll
    for (int l = 0; l < kL; ++l)
#pragma unroll
        for (int j = 0; j < kH; ++j) { hs[l][j] = 0.0f; cs[l][j] = 0.0f; }

    // ---- software-pipelined x stream: per-lane pointer, stride = one step ----
    const size_t step = (size_t)kB * (size_t)kIn;            // floats per step
    const float* xp = xin + ((size_t)base + col) * (size_t)kIn;   // t = 0
    const float* xq = xp + 16 * step;                        // prefetch cursor
    float x0 = xp[kb];                 // lo: x[n,0] ; hi: x[n,2]
    float x1 = xp[1];

    for (int t = 0; t < kT; ++t) {
        // Issue next step's loads first so their latency hides under compute.
        const float* xn = xp + ((t + 1 < kT) ? step : 0);
        float nx0 = xn[kb];
        float nx1 = xn[1];
        __builtin_prefetch(xq, 0, 3);
        xq += ((t + 16 < kT) ? step : 0);   // clamp: keep address in-bounds
        xp = xn;

        // B operand for layer 0: x^T in 4x16 layout (K=3 zero pad).
        v2f Bv;
        Bv.x = x0;
        Bv.y = lo ? x1 : 0.0f;

#pragma unroll
        for (int l = 0; l < kL; ++l) {
            // gates = W_hh * h^T + bias, then += W_ih * in^T  (chained WMMA)
            v8f acc = __builtin_amdgcn_wmma_f32_16x16x4_f32(
                false, Ahh[l], false, Bh[l], (short)0, Cb[l], false, false);
            acc = __builtin_amdgcn_wmma_f32_16x16x4_f32(
                false, Aih[l], false, Bv, (short)0, acc, false, false);

            // Low lanes hold gates 0..7 (i0-2,f0-2,g0,g1) of batch column `col`;
            // high lanes hold gates 8..11 (g2,o0-2). Swap halves to gather.
            float g2 = __shfl_xor(acc[0], 16);
            float o0 = __shfl_xor(acc[1], 16);
            float o1 = __shfl_xor(acc[2], 16);
            float o2 = __shfl_xor(acc[3], 16);

            float cn0 = sigm(acc[3]) * cs[l][0] + sigm(acc[0]) * tanh_fast(acc[6]);
            float cn1 = sigm(acc[4]) * cs[l][1] + sigm(acc[1]) * tanh_fast(acc[7]);
            float cn2 = sigm(acc[5]) * cs[l][2] + sigm(acc[2]) * tanh_fast(g2);
            float hn0 = sigm(o0) * tanh_fast(cn0);
            float hn1 = sigm(o1) * tanh_fast(cn1);
            float hn2 = sigm(o2) * tanh_fast(cn2);
            cs[l][0] = cn0; cs[l][1] = cn1; cs[l][2] = cn2;
            hs[l][0] = hn0; hs[l][1] = hn1; hs[l][2] = hn2;

            // Rebuild B operand (h^T, 4x16): one half-swap puts h2 on hi lanes.
            float h2x = __shfl_xor(hn2, 16);
            v2f nb;
            nb.x = lo ? hn0 : h2x;
            nb.y = lo ? hn1 : 0.0f;
            Bh[l] = nb;   // recurrence for next timestep
            Bv   = nb;    // feeds next layer this timestep
        }

        x0 = nx0;         // rotate the double buffer
        x1 = nx1;
    }

    // Final FC (3x3) + ReLU on last layer's hidden state; low lanes own state.
    if (lo) {
        float* op = out + (size_t)(base + col) * kH;
#pragma unroll
        for (int j = 0; j < kH; ++j) {
            float s = b_fc[j];
#pragma unroll
            for (int k = 0; k < kH; ++k) s += hs[kL - 1][k] * w_fc[j * kH + k];
            op[j] = fmaxf(s, 0.0f);
        }
    }
}

extern "C" void kernel_launch(void* const* d_in, const int* in_sizes, int n_in,
                              void* d_out, int out_size, void* d_ws, size_t ws_size,
                              hipStream_t stream) {
    const float* x   = (const float*)d_in[0];
    const float* wih = (const float*)d_in[1];
    const float* whh = (const float*)d_in[2];
    const float* bih = (const float*)d_in[3];
    const float* bhh = (const float*)d_in[4];
    const float* wfc = (const float*)d_in[5];
    const float* bfc = (const float*)d_in[6];
    float* out = (float*)d_out;

    const int wavesPerBlock = 256 / 32;                 // 8 waves
    const int tilesTotal    = kB / 16;                  // 512 wave-tiles
    dim3 block(256);
    dim3 grid(tilesTotal / wavesPerBlock);              // 64 blocks
    lstm4_wmma<<<grid, block, 0, stream>>>(x, wih, whh, bih, bhh, wfc, bfc, out);
}